// SpectralGating_27453430956348
// MI455X (gfx1250) — compile-verified
//
#include <hip/hip_runtime.h>
#include <math.h>

// SpectralGating on MI455X (gfx1250, wave32).
// out = x + irfftn((gates-1) * corner(rfftn(x)))  -- partial-DFT formulation.
// All transforms are fp32 GEMMs on V_WMMA_F32_16X16X4_F32 (16x16 tiles, K in steps of 4).
//
// Sizes: B=4, C=64, H=W=D=64. N1 = B*C*H*W = 1,048,576.
// ws float offsets:
//   B1  @ 0        [64 x 32]   forward D twiddles (cols 2kz=cos, 2kz+1=-sin; 18 used)
//   B2  @ 2048     [128 x 32]  forward W/H twiddles ([[cos,-sin],[sin,cos]] blocks)
//   B5  @ 6144     [32 x 128]  inverse H/W twiddles (e^{+i}/64)
//   B7  @ 10240    [20 x 64]   final irfft-D matrix (rows 18,19 zero pad)
//   gm1 @ 11520    [4 x 2304]  gates - 1
//   Z3  @ 32768    [36864 x 32]     corner spectrum [Re(kh)||Im(kh)]
//   Z2  @ 1212416  [36864 x 128]   (reused as T1)
//   Z1  @ 5931008  [ ... 18.9M ]    (reused as T2)

typedef float v2f __attribute__((ext_vector_type(2)));
typedef float v8f __attribute__((ext_vector_type(8)));

#define WMMA4(a, b, c) \
  __builtin_amdgcn_wmma_f32_16x16x4_f32(false, (a), false, (b), (short)0, (c), false, false)

// ---------------------------------------------------------------- twiddles --
__global__ void init_tw(float* B1, float* B2, float* B5, float* B7) {
  const float PI2 = 6.28318530717958647692f;
  const int t0 = threadIdx.x;  // one block, 256 threads
  for (int idx = t0; idx < 64 * 32; idx += 256) {
    int d = idx >> 5, j = idx & 31;
    float v = 0.f;
    if (j < 18) {
      int kz = j >> 1;
      float ang = PI2 * (float)((kz * d) & 63) * (1.0f / 64.0f);
      v = (j & 1) ? -__sinf(ang) : __cosf(ang);
    }
    B1[idx] = v;
  }
  for (int idx = t0; idx < 128 * 32; idx += 256) {
    int k = idx >> 5, j = idx & 31;
    int w = k & 63, kw = j & 15;
    float ang = PI2 * (float)((kw * w) & 63) * (1.0f / 64.0f);
    float c = __cosf(ang), s = __sinf(ang);
    B2[idx] = (k < 64) ? ((j < 16) ? c : -s) : ((j < 16) ? s : c);
  }
  for (int idx = t0; idx < 32 * 128; idx += 256) {
    int k = idx >> 7, cc = idx & 127;
    int kh = k & 15, h = cc & 63;
    float ang = PI2 * (float)((kh * h) & 63) * (1.0f / 64.0f);
    float c = __cosf(ang), s = __sinf(ang);
    float v = (k < 16) ? ((cc < 64) ? c : s) : ((cc < 64) ? -s : c);
    B5[idx] = v * (1.0f / 64.0f);
  }
  for (int idx = t0; idx < 20 * 64; idx += 256) {
    int j = idx >> 6, d = idx & 63;
    float v = 0.f;
    if (j < 18) {
      int comp = j / 9, kz = j % 9;
      float ang = PI2 * (float)((kz * d) & 63) * (1.0f / 64.0f);
      if (comp == 0) v = (kz == 0) ? (1.0f / 64.0f) : (2.0f / 64.0f) * __cosf(ang);
      else           v = (kz == 0) ? 0.f : -(2.0f / 64.0f) * __sinf(ang);
    }
    B7[idx] = v;
  }
}

// ------------------------------------------------- stage 1: forward DFT (D) --
// rows (b,c,h,w) x K=64 -> 18 cols.  Z1[b,c,h,kz,comp,w]
__global__ void __launch_bounds__(32) s1_fwd_d(const float* __restrict__ x,
                                               const float* __restrict__ B1,
                                               float* __restrict__ Z1) {
  const int lane = threadIdx.x, m = lane & 15, half = lane >> 4;
  const int r0 = blockIdx.x << 4;  // 65536 tiles
  const int w0 = r0 & 63, h = (r0 >> 6) & 63, bc = r0 >> 12;
  const float* arow = x + (size_t)(r0 + m) * 64;
  v8f acc0 = {}, acc1 = {};
#pragma unroll
  for (int t = 0; t < 16; ++t) {
    const int k0 = 4 * t + 2 * half;
    v2f a = {arow[k0], arow[k0 + 1]};
    v2f b0 = {B1[k0 * 32 + m], B1[(k0 + 1) * 32 + m]};
    v2f b1t = {B1[k0 * 32 + 16 + m], B1[(k0 + 1) * 32 + 16 + m]};
    acc0 = WMMA4(a, b0, acc0);
    acc1 = WMMA4(a, b1t, acc1);
  }
  const size_t obase = (size_t)(bc * 64 + h) * 9;
  {  // cols j = m (0..15): kz=j>>1, comp=j&1; rows (w) contiguous in memory
    float* p = Z1 + (obase + (m >> 1)) * 128 + (m & 1) * 64 + w0 + 8 * half;
    *(float4*)p       = make_float4(acc0[0], acc0[1], acc0[2], acc0[3]);
    *(float4*)(p + 4) = make_float4(acc0[4], acc0[5], acc0[6], acc0[7]);
  }
  if (m < 2) {  // cols 16,17 of second tile
    const int j = 16 + m;
    float* p = Z1 + (obase + (j >> 1)) * 128 + (j & 1) * 64 + w0 + 8 * half;
    *(float4*)p       = make_float4(acc1[0], acc1[1], acc1[2], acc1[3]);
    *(float4*)(p + 4) = make_float4(acc1[4], acc1[5], acc1[6], acc1[7]);
  }
}

// ------------------------------------------------- stage 2: forward DFT (W) --
// rows (b,c,kz,h) x K=128 -> 32 cols.  Z2[b,c,kw,kz,comp,h]
__global__ void __launch_bounds__(32) s2_fwd_w(const float* __restrict__ Z1,
                                               const float* __restrict__ B2,
                                               float* __restrict__ Z2) {
  const int lane = threadIdx.x, m = lane & 15, half = lane >> 4;
  const int r0 = blockIdx.x << 4;  // 9216 tiles
  const int h0 = r0 & 63;
  const int tmp = r0 >> 6, kz = tmp % 9, bc = tmp / 9;
  const float* arow = Z1 + ((size_t)(bc * 64 + h0 + m) * 9 + kz) * 128;
  v8f acc0 = {}, acc1 = {};
#pragma unroll
  for (int t = 0; t < 32; ++t) {
    const int k0 = 4 * t + 2 * half;
    v2f a = {arow[k0], arow[k0 + 1]};
    v2f b0 = {B2[k0 * 32 + m], B2[(k0 + 1) * 32 + m]};
    v2f b1t = {B2[k0 * 32 + 16 + m], B2[(k0 + 1) * 32 + 16 + m]};
    acc0 = WMMA4(a, b0, acc0);
    acc1 = WMMA4(a, b1t, acc1);
  }
  {  // col m -> (kw=m, comp=0); rows (h) contiguous
    float* p = Z2 + ((size_t)((bc * 16 + m) * 9 + kz) * 2 + 0) * 64 + h0 + 8 * half;
    *(float4*)p       = make_float4(acc0[0], acc0[1], acc0[2], acc0[3]);
    *(float4*)(p + 4) = make_float4(acc0[4], acc0[5], acc0[6], acc0[7]);
  }
  {  // col 16+m -> (kw=m, comp=1)
    float* p = Z2 + ((size_t)((bc * 16 + m) * 9 + kz) * 2 + 1) * 64 + h0 + 8 * half;
    *(float4*)p       = make_float4(acc1[0], acc1[1], acc1[2], acc1[3]);
    *(float4*)(p + 4) = make_float4(acc1[4], acc1[5], acc1[6], acc1[7]);
  }
}

// ------------------------------------------------- stage 3: forward DFT (H) --
// rows n3=(b,c,kw,kz) x K=128 -> 32 cols.  Z3[n3][32] = [Re(kh)||Im(kh)]
__global__ void __launch_bounds__(32) s3_fwd_h(const float* __restrict__ Z2,
                                               const float* __restrict__ B2,
                                               float* __restrict__ Z3) {
  const int lane = threadIdx.x, m = lane & 15, half = lane >> 4;
  const int r0 = blockIdx.x << 4;  // 2304 tiles
  const float* arow = Z2 + (size_t)(r0 + m) * 128;
  v8f acc0 = {}, acc1 = {};
#pragma unroll
  for (int t = 0; t < 32; ++t) {
    const int k0 = 4 * t + 2 * half;
    v2f a = {arow[k0], arow[k0 + 1]};
    v2f b0 = {B2[k0 * 32 + m], B2[(k0 + 1) * 32 + m]};
    v2f b1t = {B2[k0 * 32 + 16 + m], B2[(k0 + 1) * 32 + 16 + m]};
    acc0 = WMMA4(a, b0, acc0);
    acc1 = WMMA4(a, b1t, acc1);
  }
#pragma unroll
  for (int v = 0; v < 8; ++v) {  // direct scatter; tiny (4.7 MB)
    Z3[(size_t)(r0 + v + 8 * half) * 32 + m]      = acc0[v];
    Z3[(size_t)(r0 + v + 8 * half) * 32 + 16 + m] = acc1[v];
  }
}

// ------------------------------------------------------- stage 4a: gate MLP --
__global__ void mlp_kernel(const float* __restrict__ Z3, const float* __restrict__ w1,
                           const float* __restrict__ b1, const float* __restrict__ w2,
                           const float* __restrict__ b2, float* __restrict__ gm1) {
  __shared__ float pooled[256];
  __shared__ float hbuf[64];
  const int t = threadIdx.x;
  // pooled[b,c] = Re(X[0,0,0]) / (H*W*D); Z3 row bc*144, element 0
  pooled[t] = Z3[(size_t)t * 144 * 32] * (1.0f / 262144.0f);
  __syncthreads();
  if (t < 64) {
    const int b = t >> 4, i = t & 15;
    float s = b1[i];
    for (int c = 0; c < 64; ++c) s += pooled[b * 64 + c] * w1[c * 16 + i];
    hbuf[t] = s > 0.f ? s : 0.f;
  }
  __syncthreads();
  for (int idx = t; idx < 4 * 2304; idx += 256) {
    const int b = idx / 2304, col = idx % 2304;
    float s = b2[col];
    for (int i = 0; i < 16; ++i) s += hbuf[b * 16 + i] * w2[i * 2304 + col];
    gm1[idx] = 1.0f / (1.0f + __expf(-s)) - 1.0f;
  }
}

// --------------------------------------------- stage 4b: apply (g-1) in place
__global__ void gate_kernel(float* __restrict__ Z3, const float* __restrict__ gm1) {
  const int idx = blockIdx.x * 256 + threadIdx.x;  // 1,179,648 exact
  const int j = idx & 31, kh = j & 15;
  const int n3 = idx >> 5;
  const int kz = n3 % 9, t2 = n3 / 9, kw = t2 & 15, bc = t2 >> 4, b = bc >> 6;
  Z3[idx] *= gm1[(size_t)b * 2304 + (kh * 16 + kw) * 9 + kz];
}

// ------------------------------------------------- stage 5: inverse DFT (H) --
// rows (b,c,kz,kw) x K=32 -> 128 cols.  T1[b,c,h,kz,comp,kw]
__global__ void __launch_bounds__(32) s5_inv_h(const float* __restrict__ Cg,
                                               const float* __restrict__ B5,
                                               float* __restrict__ T1) {
  const int lane = threadIdx.x, m = lane & 15, half = lane >> 4;
  const int tile = blockIdx.x;  // 2304: (bc,kz), rows = kw 0..15
  const int kz = tile % 9, bc = tile / 9;
  const float* arow = Cg + ((size_t)(bc * 16 + m) * 9 + kz) * 32;
  v8f acc[8] = {};
#pragma unroll
  for (int t = 0; t < 8; ++t) {
    const int k0 = 4 * t + 2 * half;
    v2f a = {arow[k0], arow[k0 + 1]};
#pragma unroll
    for (int ct = 0; ct < 8; ++ct) {
      v2f b = {B5[k0 * 128 + ct * 16 + m], B5[(k0 + 1) * 128 + ct * 16 + m]};
      acc[ct] = WMMA4(a, b, acc[ct]);
    }
  }
#pragma unroll
  for (int ct = 0; ct < 8; ++ct) {  // col cc=ct*16+m: comp=cc>>6, h=cc&63; rows(kw) contiguous
    const int comp = ct >> 2, hval = (ct & 3) * 16 + m;
    float* p = T1 + ((size_t)((bc * 64 + hval) * 9 + kz) * 2 + comp) * 16 + 8 * half;
    *(float4*)p       = make_float4(acc[ct][0], acc[ct][1], acc[ct][2], acc[ct][3]);
    *(float4*)(p + 4) = make_float4(acc[ct][4], acc[ct][5], acc[ct][6], acc[ct][7]);
  }
}

// ------------------------------------------------- stage 6: inverse DFT (W) --
// rows (b,c,kz,h) x K=32 -> 128 cols.  T2[b,c,h,j,w], j = comp*9+kz
__global__ void __launch_bounds__(32) s6_inv_w(const float* __restrict__ T1,
                                               const float* __restrict__ B5,
                                               float* __restrict__ T2) {
  __shared__ float lds[16 * 132];
  const int lane = threadIdx.x, m = lane & 15, half = lane >> 4;
  const int r0 = blockIdx.x << 4;  // 9216 tiles
  const int h0 = r0 & 63;
  const int tmp = r0 >> 6, kz = tmp % 9, bc = tmp / 9;
  const float* arow = T1 + ((size_t)(bc * 64 + h0 + m) * 9 + kz) * 32;
  v8f acc[8] = {};
#pragma unroll
  for (int t = 0; t < 8; ++t) {
    const int k0 = 4 * t + 2 * half;
    v2f a = {arow[k0], arow[k0 + 1]};
#pragma unroll
    for (int ct = 0; ct < 8; ++ct) {
      v2f b = {B5[k0 * 128 + ct * 16 + m], B5[(k0 + 1) * 128 + ct * 16 + m]};
      acc[ct] = WMMA4(a, b, acc[ct]);
    }
  }
#pragma unroll
  for (int ct = 0; ct < 8; ++ct)
#pragma unroll
    for (int v = 0; v < 8; ++v) lds[(v + 8 * half) * 132 + ct * 16 + m] = acc[ct][v];
  __syncthreads();
  // lane l writes row=(l&15), comp=(l>>4): 64 contiguous floats along w
  const float* src = lds + m * 132 + half * 64;
  float* p = T2 + ((size_t)(bc * 64 + h0 + m) * 18 + half * 9 + kz) * 64;
#pragma unroll
  for (int q = 0; q < 16; ++q) *(float4*)(p + 4 * q) = *(const float4*)(src + 4 * q);
}

// -------------------------------- stage 7: irfft (D) fused with out = x + d --
// per tile: A = 16 w-rows x K=20(18) from T2[b,c,h,*,w], B7[20x64] -> delta[16x64]
__global__ void __launch_bounds__(32) s7_final(const float* __restrict__ T2,
                                               const float* __restrict__ B7,
                                               const float* __restrict__ x,
                                               float* __restrict__ out) {
  __shared__ float lds[16 * 68];
  const int lane = threadIdx.x, m = lane & 15, half = lane >> 4;
  const int tile = blockIdx.x;  // 65536: (bch, wblk)
  const int wblk = tile & 3, bch = tile >> 2;
  const int w0 = wblk * 16;
  const float* Abase = T2 + (size_t)bch * 18 * 64 + w0;  // + j*64 + m
  v8f acc[4] = {};
#pragma unroll
  for (int t = 0; t < 5; ++t) {
    const int k0 = 4 * t + 2 * half;
    v2f a;
    a.x = (k0 < 18)     ? Abase[k0 * 64 + m]       : 0.f;
    a.y = (k0 + 1 < 18) ? Abase[(k0 + 1) * 64 + m] : 0.f;
#pragma unroll
    for (int ct = 0; ct < 4; ++ct) {  // B7 rows 18,19 are zero-filled
      v2f b = {B7[k0 * 64 + ct * 16 + m], B7[(k0 + 1) * 64 + ct * 16 + m]};
      acc[ct] = WMMA4(a, b, acc[ct]);
    }
  }
#pragma unroll
  for (int ct = 0; ct < 4; ++ct)
#pragma unroll
    for (int v = 0; v < 8; ++v) lds[(v + 8 * half) * 68 + ct * 16 + m] = acc[ct][v];
  __syncthreads();
  const int row = lane >> 1, dseg = (lane & 1) * 32;
  const size_t rb = ((size_t)bch * 64 + w0 + row) * 64 + dseg;
  const float* src = lds + row * 68 + dseg;
#pragma unroll
  for (int q = 0; q < 8; ++q) {
    float4 d4 = *(const float4*)(src + 4 * q);
    float4 xv = *(const float4*)(x + rb + 4 * q);
    *(float4*)(out + rb + 4 * q) =
        make_float4(xv.x + d4.x, xv.y + d4.y, xv.z + d4.z, xv.w + d4.w);
  }
}

// ----------------------------------------------------------------- launcher --
extern "C" void kernel_launch(void* const* d_in, const int* in_sizes, int n_in,
                              void* d_out, int out_size, void* d_ws, size_t ws_size,
                              hipStream_t stream) {
  const float* x  = (const float*)d_in[0];
  const float* w1 = (const float*)d_in[1];
  const float* b1 = (const float*)d_in[2];
  const float* w2 = (const float*)d_in[3];
  const float* b2 = (const float*)d_in[4];
  float* out = (float*)d_out;
  float* ws = (float*)d_ws;

  float* B1  = ws + 0;
  float* B2  = ws + 2048;
  float* B5  = ws + 6144;
  float* B7  = ws + 10240;
  float* gm1 = ws + 11520;
  float* Z3  = ws + 32768;     // 1,179,648 floats
  float* Z2  = ws + 1212416;   // 4,718,592 floats (reused as T1)
  float* Z1  = ws + 5931008;   // 18,874,368 floats (reused as T2)

  init_tw<<<1, 256, 0, stream>>>(B1, B2, B5, B7);
  s1_fwd_d<<<65536, 32, 0, stream>>>(x, B1, Z1);
  s2_fwd_w<<<9216, 32, 0, stream>>>(Z1, B2, Z2);
  s3_fwd_h<<<2304, 32, 0, stream>>>(Z2, B2, Z3);
  mlp_kernel<<<1, 256, 0, stream>>>(Z3, w1, b1, w2, b2, gm1);
  gate_kernel<<<4608, 256, 0, stream>>>(Z3, gm1);
  s5_inv_h<<<2304, 32, 0, stream>>>(Z3, B5, /*T1=*/Z2);
  s6_inv_w<<<9216, 32, 0, stream>>>(/*T1=*/Z2, B5, /*T2=*/Z1);
  s7_final<<<65536, 32, 0, stream>>>(/*T2=*/Z1, B7, x, out);
}